// HardVoxelization_40235253629492
// MI455X (gfx1250) — compile-verified
//
#include <hip/hip_runtime.h>
#include <hip/hip_bf16.h>
#include <stdint.h>

// ---------------- problem constants (match reference) ----------------
#define N_GX 256
#define N_GY 256
#define N_GZ 1
#define NVOX 65536          // GX*GY*GZ
#define MAX_VOX 16000
#define MAX_PTS 32
#define DONE_SENTINEL 0x7FFFFFFE
#define IMAX 0x7FFFFFFF

// output layout (floats)
#define VOX_OFF   0                         // 16000*32*4 = 2048000
#define COORD_OFF 2048000                   // 16000*3   = 48000
#define NUMP_OFF  2096000                   // 16000

typedef unsigned int u32x4 __attribute__((ext_vector_type(4)));
typedef int          i32x8 __attribute__((ext_vector_type(8)));
typedef int          i32x4 __attribute__((ext_vector_type(4)));

// ---------------- K0: zero outputs + init tables ----------------
__global__ void k_init(float* __restrict__ out, int out_n,
                       int* __restrict__ cnt, int* __restrict__ lastSel,
                       int* __restrict__ minIdx) {
  int i = blockIdx.x * blockDim.x + threadIdx.x;
  if (i < out_n) out[i] = 0.0f;
  if (i < NVOX) {
    cnt[i]     = 0;
    lastSel[i] = -1;
    minIdx[i]  = IMAX;
  }
}

// ---------------- K1: hash per point + per-voxel histogram ----------------
__global__ void k_hash(const float4* __restrict__ pts, int* __restrict__ hash,
                       int* __restrict__ cnt, int n) {
  int i = blockIdx.x * blockDim.x + threadIdx.x;
  if (i >= n) return;
  float4 p = pts[i];
  const float lox = 0.0f,  loy = -25.6f, loz = -3.0f;
  const float hix = 51.2f, hiy = 25.6f,  hiz = 5.0f;
  const float vx = 0.2f,   vy = 0.2f,    vz = 8.0f;
  bool valid = (p.x >= lox) && (p.x < hix) &&
               (p.y >= loy) && (p.y < hiy) &&
               (p.z >= loz) && (p.z < hiz);
  int h = NVOX;  // invalid marker (== BIG)
  if (valid) {
    int cx = (int)floorf((p.x - lox) / vx);
    int cy = (int)floorf((p.y - loy) / vy);
    int cz = (int)floorf((p.z - loz) / vz);
    cx = cx < 0 ? 0 : (cx > N_GX - 1 ? N_GX - 1 : cx);
    cy = cy < 0 ? 0 : (cy > N_GY - 1 ? N_GY - 1 : cy);
    cz = cz < 0 ? 0 : (cz > N_GZ - 1 ? N_GZ - 1 : cz);
    h = cx * (N_GY * N_GZ) + cy * N_GZ + cz;
    atomicAdd(&cnt[h], 1);
  }
  hash[i] = h;
}

// ---------------- K2: occupancy scan -> vid, coords, num_points ----------
// Single workgroup, 1024 threads. Counts staged into LDS via the CDNA5
// Tensor Data Mover (tensor_load_to_lds + s_wait_tensorcnt), chunked 8192
// ints (32 KB) at a time as a 1024x8 2-D tile.
#define SCAN_T 1024
#define CHUNK  8192
#define NCHUNK (NVOX / CHUNK)

__global__ __launch_bounds__(SCAN_T) void k_scan(const int* __restrict__ cnt,
                                                 int* __restrict__ vid,
                                                 float* __restrict__ coords_out,
                                                 float* __restrict__ nump_out) {
  __shared__ int s_cnt[CHUNK];
  __shared__ int s_sum[SCAN_T];
  const int t = threadIdx.x;
  int running = 0;
  for (int c = 0; c < NCHUNK; ++c) {
    const int base = c * CHUNK;
#if __has_builtin(__builtin_amdgcn_tensor_load_to_lds) && __has_builtin(__builtin_amdgcn_s_wait_tensorcnt)
    if (t < 32) {  // wave 0 issues the TDM descriptor once
      unsigned long long gaddr = (unsigned long long)(const void*)(cnt + base);
      unsigned ldsa = (unsigned)(unsigned long long)(const void*)&s_cnt[0];
      u32x4 g0;
      g0[0] = 1u;                                      // count=1, user mode
      g0[1] = ldsa;                                    // lds_addr (bytes)
      g0[2] = (unsigned)(gaddr & 0xFFFFFFFFu);         // global_addr[31:0]
      g0[3] = (unsigned)((gaddr >> 32) & 0x01FFFFFFu)  // global_addr[56:32]
              | (2u << 30);                            // type = 2 ("image")
      i32x8 g1;
      g1[0] = (int)(2u << 16);       // workgroup_mask=0, data_size=2 (4B)
      g1[1] = (int)(1024u << 16);    // tensor_dim0 lo16 = 1024 (bits 63:48)
      g1[2] = (int)(8u << 16);       // tensor_dim0 hi=0; tensor_dim1 lo16 = 8
      g1[3] = (int)(1024u << 16);    // tensor_dim1 hi=0; tile_dim0 = 1024
      g1[4] = 8;                     // tile_dim1 = 8, tile_dim2 = 0
      g1[5] = 1024;                  // tensor_dim0_stride lo32 = 1024 elems
      g1[6] = 0;                     // stride0 hi, stride1 lo
      g1[7] = 0;                     // stride1 hi
      i32x4 z4 = {0, 0, 0, 0};
#if __clang_major__ >= 23
      i32x8 z8 = {0, 0, 0, 0, 0, 0, 0, 0};
      __builtin_amdgcn_tensor_load_to_lds(g0, g1, z4, z4, z8, 0);
#else
      __builtin_amdgcn_tensor_load_to_lds(g0, g1, z4, z4, 0);
#endif
      __builtin_amdgcn_s_wait_tensorcnt(0);
    }
#else
    for (int j = t; j < CHUNK; j += SCAN_T) s_cnt[j] = cnt[base + j];
#endif
    __syncthreads();

    // thread-local prefix over 8 occupancy flags
    int local[8];
    int tsum = 0;
#pragma unroll
    for (int j = 0; j < 8; ++j) {
      int cc = s_cnt[t * 8 + j];
      local[j] = tsum;
      tsum += (cc > 0) ? 1 : 0;
    }
    s_sum[t] = tsum;
    __syncthreads();
    // block-wide inclusive scan (Hillis-Steele)
    for (int off = 1; off < SCAN_T; off <<= 1) {
      int v = s_sum[t];
      int add = (t >= off) ? s_sum[t - off] : 0;
      __syncthreads();
      s_sum[t] = v + add;
      __syncthreads();
    }
    int excl = s_sum[t] - tsum;
    int chunkTotal = s_sum[SCAN_T - 1];

#pragma unroll
    for (int j = 0; j < 8; ++j) {
      int v = base + t * 8 + j;
      int cc = s_cnt[t * 8 + j];
      if (cc > 0) {
        int g = running + excl + local[j];
        vid[v] = g;
        if (g < MAX_VOX) {
          int cz = v % N_GZ;
          int cy = (v / N_GZ) % N_GY;
          int cx = v / (N_GZ * N_GY);
          coords_out[(size_t)g * 3 + 0] = (float)cx;
          coords_out[(size_t)g * 3 + 1] = (float)cy;
          coords_out[(size_t)g * 3 + 2] = (float)cz;
          nump_out[g] = (float)(cc < MAX_PTS ? cc : MAX_PTS);
        }
      } else {
        vid[v] = -1;
      }
    }
    running += chunkTotal;
    __syncthreads();
  }
}

// ---------------- K3: one selection round (atomicMin tournament) ---------
// 4-wide: one global_load_b128 of hashes per thread; table probes are
// L2-resident and heavily filtered, so this kernel is issue-limited and the
// 4x vectorization quarters instruction/wave-launch cost per round.
__global__ void k_round(const int4* __restrict__ hash4,
                        const int* __restrict__ lastSel,
                        int* __restrict__ minIdx, int n4, int n) {
  int t = blockIdx.x * blockDim.x + threadIdx.x;
  if (t >= n4) return;
  int pf = t + 4096;
  if (pf < n4) __builtin_prefetch((const int*)(hash4 + pf), 0, 1); // global_prefetch_b8
  int4 h4 = hash4[t];
  int base = t * 4;
#pragma unroll
  for (int j = 0; j < 4; ++j) {
    int h = (j == 0) ? h4.x : (j == 1) ? h4.y : (j == 2) ? h4.z : h4.w;
    int i = base + j;
    if (i < n && h < NVOX) {
      int ls = lastSel[h];                 // L2-resident 256KB table
      if (i > ls) {
        // pre-read filter: minIdx only decreases, so skipping when it is
        // already <= i is safe and kills most atomic traffic
        if (i < minIdx[h]) atomicMin(&minIdx[h], i);
      }
    }
  }
}

// ---------------- K4: consume round winner, emit voxel row r -------------
__global__ void k_update(int* __restrict__ lastSel, int* __restrict__ minIdx,
                         const int* __restrict__ vid,
                         const float4* __restrict__ pts,
                         float* __restrict__ voxels_out, int r) {
  int v = blockIdx.x * blockDim.x + threadIdx.x;
  if (v >= NVOX) return;
  int m = minIdx[v];
  if (m != IMAX) {
    lastSel[v] = m;           // m is now selected; all idx<=m are selected
    minIdx[v]  = IMAX;
    int id = vid[v];
    if (id >= 0 && id < MAX_VOX) {
      float4 p = pts[m];
      float4* dst = (float4*)(voxels_out + ((size_t)id * MAX_PTS + r) * 4);
      *dst = p;               // global_store_b128
    }
  } else {
    lastSel[v] = DONE_SENTINEL;  // exhausted voxel: its points skip fast
  }
}

// ---------------- host launcher ----------------
extern "C" void kernel_launch(void* const* d_in, const int* in_sizes, int n_in,
                              void* d_out, int out_size, void* d_ws, size_t ws_size,
                              hipStream_t stream) {
  const float* points = (const float*)d_in[0];
  const int N = in_sizes[0] / 4;          // 4,000,000 points x 4 feats
  const int N4 = (N + 3) / 4;

  float* out    = (float*)d_out;
  float* voxels = out + VOX_OFF;
  float* coords = out + COORD_OFF;
  float* nump   = out + NUMP_OFF;

  int* ws      = (int*)d_ws;
  int* hash    = ws;                 // N ints (N4*4 capacity assumed)
  int* cnt     = ws + ((N + 3) & ~3);// NVOX, keep 16B alignment for hash4
  int* vid     = cnt + NVOX;         // NVOX
  int* lastSel = vid + NVOX;         // NVOX
  int* minIdx  = lastSel + NVOX;     // NVOX

  const int TB = 256;
  int initN = out_size > NVOX ? out_size : NVOX;
  k_init<<<(initN + TB - 1) / TB, TB, 0, stream>>>(out, out_size, cnt, lastSel, minIdx);
  k_hash<<<(N + TB - 1) / TB, TB, 0, stream>>>((const float4*)points, hash, cnt, N);
  k_scan<<<1, SCAN_T, 0, stream>>>(cnt, vid, coords, nump);
  for (int r = 0; r < MAX_PTS; ++r) {
    k_round<<<(N4 + TB - 1) / TB, TB, 0, stream>>>((const int4*)hash, lastSel, minIdx, N4, N);
    k_update<<<(NVOX + TB - 1) / TB, TB, 0, stream>>>(lastSel, minIdx, vid,
                                                      (const float4*)points, voxels, r);
  }
}